// SlotContrastiveLoss_82471962018318
// MI455X (gfx1250) — compile-verified
//
#include <hip/hip_runtime.h>

typedef __attribute__((ext_vector_type(2))) float v2f;
typedef __attribute__((ext_vector_type(8))) float v8f;

#define K_SLOTS 8
#define D_DIM   256

// One wave32 handles TWO batch samples (16 flat rows of [B*K, D]).
// WMMA f32 16x16x4 accumulates the packed 16x16 Gram; diagonal 8x8 blocks
// are the per-sample logits (pre-scale).
__global__ __launch_bounds__(256) void slot_contrastive_kernel(
    const float* __restrict__ drone, const float* __restrict__ sat,
    const float* __restrict__ temp_ptr, float* __restrict__ partials)
{
    const int tid       = threadIdx.x;
    const int lane      = tid & 31;
    const int waveInBlk = tid >> 5;
    const int gwave     = (int)((blockIdx.x * blockDim.x + tid) >> 5);

    // tile row m (0..15)  <->  flat row 16*gwave + m  (2 contiguous samples)
    const int   row  = (gwave << 4) + (lane & 15);
    const int   h    = lane >> 4;              // lane half: K-slots {0,1} vs {2,3}
    const float invT = 1.0f / temp_ptr[0];

    const float4* dsr = (const float4*)(drone + (size_t)row * D_DIM);
    const float4* ssr = (const float4*)(sat   + (size_t)row * D_DIM);

    v8f c = {0.f, 0.f, 0.f, 0.f, 0.f, 0.f, 0.f, 0.f};
    float nds = 0.f, nss = 0.f;

    // Iteration t covers D[8t..8t+7]; this lane's half loads 4 of them (b128).
    #pragma unroll 4
    for (int t = 0; t < D_DIM / 8; ++t) {
        float4 a4 = dsr[2 * t + h];
        float4 b4 = ssr[2 * t + h];
        nds += a4.x * a4.x + a4.y * a4.y + a4.z * a4.z + a4.w * a4.w;
        nss += b4.x * b4.x + b4.y * b4.y + b4.z * b4.z + b4.w * b4.w;
        v2f a0 = {a4.x, a4.y}, b0 = {b4.x, b4.y};
        v2f a1 = {a4.z, a4.w}, b1 = {b4.z, b4.w};
        c = __builtin_amdgcn_wmma_f32_16x16x4_f32(false, a0, false, b0,
                                                  (short)0, c, false, false);
        c = __builtin_amdgcn_wmma_f32_16x16x4_f32(false, a1, false, b1,
                                                  (short)0, c, false, false);
    }

    // Lanes r and r+16 each hold half of row r's squared norm: combine.
    nds += __shfl_xor(nds, 16, 32);
    nss += __shfl_xor(nss, 16, 32);
    const float inv_ss_own = 1.0f / fmaxf(sqrtf(nss), 1e-12f);

    // Valid output lanes: 0-7 = sample 2w (rows/cols 0-7 of tile),
    //                     24-31 = sample 2w+1 (rows/cols 8-15 of tile).
    const bool valid0  = (lane < 8);
    const bool valid1  = (lane >= 24);
    const bool valid   = valid0 || valid1;
    const int  rowbase = valid1 ? 8 : 0;
    const int  jloc    = valid0 ? lane : (lane - 24);

    float acc  = 0.f;   // (1/8) * sum_k lse_k  (identical across the 8-lane group)
    float diag = 0.f;   // this column's diagonal logit

    #pragma unroll
    for (int k = 0; k < 8; ++k) {
        float ndk    = __shfl(nds, rowbase + k, 32);
        float inv_dk = 1.0f / fmaxf(sqrtf(ndk), 1e-12f);
        float logit  = c[k] * inv_dk * inv_ss_own * invT;

        // logsumexp over the 8 columns j (xor butterfly stays inside the group)
        float m = logit;
        m = fmaxf(m, __shfl_xor(m, 1, 32));
        m = fmaxf(m, __shfl_xor(m, 2, 32));
        m = fmaxf(m, __shfl_xor(m, 4, 32));
        float e = expf(logit - m);
        e += __shfl_xor(e, 1, 32);
        e += __shfl_xor(e, 2, 32);
        e += __shfl_xor(e, 4, 32);
        float lse = m + logf(e);

        acc += 0.125f * lse;
        if (k == jloc) diag = logit;
    }

    // Per-lane contribution; group-sum over 8 lanes = sum_k (lse_k - pos_k).
    float v = valid ? (acc - diag) : 0.f;

    // Full-wave deterministic butterfly sum.
    v += __shfl_xor(v, 1, 32);
    v += __shfl_xor(v, 2, 32);
    v += __shfl_xor(v, 4, 32);
    v += __shfl_xor(v, 8, 32);
    v += __shfl_xor(v, 16, 32);

    __shared__ float wsum[8];
    if (lane == 0) wsum[waveInBlk] = v;
    __syncthreads();
    if (tid == 0) {
        float s = 0.f;
        #pragma unroll
        for (int i = 0; i < 8; ++i) s += wsum[i];
        partials[blockIdx.x] = s;
    }
}

// Deterministic fixed-order final reduction: loss = scale * sum(partials),
// scale = 1 / (K * B)  (mean over k, then mean over batch).
__global__ __launch_bounds__(256) void slot_reduce_kernel(
    const float* __restrict__ partials, float* __restrict__ out,
    int n, float scale)
{
    __shared__ float sm[256];
    float s = 0.f;
    for (int i = threadIdx.x; i < n; i += 256) s += partials[i];
    sm[threadIdx.x] = s;
    __syncthreads();
    for (int w = 128; w > 0; w >>= 1) {
        if ((int)threadIdx.x < w) sm[threadIdx.x] += sm[threadIdx.x + w];
        __syncthreads();
    }
    if (threadIdx.x == 0) out[0] = sm[0] * scale;
}

extern "C" void kernel_launch(void* const* d_in, const int* in_sizes, int n_in,
                              void* d_out, int out_size, void* d_ws, size_t ws_size,
                              hipStream_t stream) {
    const float* drone = (const float*)d_in[0];   // [B, K, D] f32
    const float* sat   = (const float*)d_in[1];   // [B, K, D] f32
    // d_in[2] = labels (unused by the reference math)
    const float* temp  = (const float*)d_in[3];   // scalar f32

    const int B = in_sizes[0] / (K_SLOTS * D_DIM);      // 32768
    const int nblocks = B / 16;                         // 16 samples per block
    float* partials = (float*)d_ws;                     // nblocks floats

    slot_contrastive_kernel<<<nblocks, 256, 0, stream>>>(drone, sat, temp, partials);

    const float scale = 1.0f / (8.0f * (float)B);
    slot_reduce_kernel<<<1, 256, 0, stream>>>(partials, (float*)d_out, nblocks, scale);
}